// ParticleModel_54494545052025
// MI455X (gfx1250) — compile-verified
//
#include <hip/hip_runtime.h>

// CDNA5 / gfx1250: wave32, WMMA 16x16x4 fp32
typedef __attribute__((ext_vector_type(2))) float v2f;
typedef __attribute__((ext_vector_type(8))) float v8f;

#define DIM_B 4096
#define DIM_M 128
#define DIM_C 128
#define DIM_Q 256
#define BPB   16    // batches (M-tile rows) per block in kernel 1

// ---------------------------------------------------------------------------
// Kernel 1: per-block (16 batches):
//   xsum  = masked row-sum of xss            (memory-bound streaming)
//   ctx   = xsum @ Wk^T      (16x256x128)    WMMA f32 16x16x4
//   cs    = ctx  @ Wc^T      (16x128x128)    WMMA f32 16x16x4
//   v     = cs   @ proj      (16x128x256)    WMMA f32 16x16x4  -> workspace
// ---------------------------------------------------------------------------
__global__ __launch_bounds__(256) void ragged_pool_gemm_kernel(
    const float* __restrict__ xss,
    const float* __restrict__ Wk,
    const float* __restrict__ Wc,
    const float* __restrict__ proj,
    const int*   __restrict__ seq_lengths,
    float*       __restrict__ v_out)
{
    __shared__ float s_xsum[BPB][DIM_Q + 1];  // 16 x 257
    __shared__ float s_ctx [BPB][DIM_C + 1];  // 16 x 129
    __shared__ float s_cs  [BPB][DIM_C + 1];  // 16 x 129

    const int t    = threadIdx.x;      // 0..255 == column q
    const int b0   = blockIdx.x * BPB;
    const int wave = t >> 5;           // 0..7
    const int lane = t & 31;
    const int ln   = lane & 15;        // lane within half-wave
    const int hi   = lane >> 4;        // 0: lanes 0-15, 1: lanes 16-31

    // ---- Phase 0: masked column sums (only valid rows are read) ----
    for (int i = 0; i < BPB; ++i) {
        const int len = seq_lengths[b0 + i];
        const float* xb = xss + (size_t)(b0 + i) * DIM_M * DIM_Q + t;
        float acc = 0.0f;
        int m = 0;
        for (; m + 4 <= len; m += 4) {
            acc += xb[(m + 0) * DIM_Q];
            acc += xb[(m + 1) * DIM_Q];
            acc += xb[(m + 2) * DIM_Q];
            acc += xb[(m + 3) * DIM_Q];
        }
        for (; m < len; ++m) acc += xb[m * DIM_Q];
        s_xsum[i][t] = acc;
    }
    __syncthreads();

    // ---- GEMM 1: ctx(16x128) = xsum(16x256) @ Wk^T ; wave w owns N-tile w --
    {
        const int n0 = wave * 16;
        v8f acc = {};
        for (int k0 = 0; k0 < DIM_Q; k0 += 4) {
            const int ka = k0 + 2 * hi;          // ISA 16x4 A layout: v0=K{0,2}, v1=K{1,3}
            v2f a, bf;
            a.x  = s_xsum[ln][ka];
            a.y  = s_xsum[ln][ka + 1];
            const float* wrow = Wk + (size_t)(n0 + ln) * DIM_Q;  // B[k][n] = Wk[n][k]
            bf.x = wrow[ka];
            bf.y = wrow[ka + 1];
            acc = __builtin_amdgcn_wmma_f32_16x16x4_f32(
                    false, a, false, bf, (short)0, acc, false, false);
        }
        for (int r = 0; r < 8; ++r)              // D: VGPR r = rows r / r+8
            s_ctx[r + 8 * hi][n0 + ln] = acc[r];
    }
    __syncthreads();

    // ---- GEMM 2: cs(16x128) = ctx(16x128) @ Wc^T --------------------------
    {
        const int n0 = wave * 16;
        v8f acc = {};
        for (int k0 = 0; k0 < DIM_C; k0 += 4) {
            const int ka = k0 + 2 * hi;
            v2f a, bf;
            a.x  = s_ctx[ln][ka];
            a.y  = s_ctx[ln][ka + 1];
            const float* wrow = Wc + (size_t)(n0 + ln) * DIM_C;  // B[k][n] = Wc[n][k]
            bf.x = wrow[ka];
            bf.y = wrow[ka + 1];
            acc = __builtin_amdgcn_wmma_f32_16x16x4_f32(
                    false, a, false, bf, (short)0, acc, false, false);
        }
        for (int r = 0; r < 8; ++r)
            s_cs[r + 8 * hi][n0 + ln] = acc[r];
    }
    __syncthreads();

    // ---- GEMM 3: v(16x256) = cs(16x128) @ proj ; wave owns tiles w, w+8 ---
    for (int half = 0; half < 2; ++half) {
        const int n0 = (wave + 8 * half) * 16;
        v8f acc = {};
        for (int k0 = 0; k0 < DIM_C; k0 += 4) {
            const int ka = k0 + 2 * hi;
            v2f a, bf;
            a.x  = s_cs[ln][ka];
            a.y  = s_cs[ln][ka + 1];
            bf.x = proj[(size_t)(ka)     * DIM_Q + n0 + ln];     // B[k][n] = proj[k][n]
            bf.y = proj[(size_t)(ka + 1) * DIM_Q + n0 + ln];
            acc = __builtin_amdgcn_wmma_f32_16x16x4_f32(
                    false, a, false, bf, (short)0, acc, false, false);
        }
        for (int r = 0; r < 8; ++r)
            v_out[(size_t)(b0 + r + 8 * hi) * DIM_Q + n0 + ln] = acc[r];
    }
}

// ---------------------------------------------------------------------------
// Kernel 2: one wave32 per valid token t:
//   ys[t] = dot(v[b], xss[b,m,:]),  b*128+m = gather_idx[t]
// ---------------------------------------------------------------------------
__global__ __launch_bounds__(256) void token_dot_kernel(
    const float* __restrict__ xss,
    const float* __restrict__ v_ws,
    const int*   __restrict__ gather_idx,
    float*       __restrict__ ys,
    int total)
{
    const int wave = (int)((blockIdx.x * blockDim.x + threadIdx.x) >> 5);
    const int lane = threadIdx.x & 31;
    if (wave >= total) return;

    const int idx = gather_idx[wave];          // b*128 + m
    const int b   = idx >> 7;

    const float4* xrow = (const float4*)(xss + (size_t)idx * DIM_Q);
    const float4* vrow = (const float4*)(v_ws + (size_t)b * DIM_Q);

    const float4 x0 = xrow[lane];
    const float4 x1 = xrow[lane + 32];
    const float4 v0 = vrow[lane];
    const float4 v1 = vrow[lane + 32];

    float s = x0.x * v0.x + x0.y * v0.y + x0.z * v0.z + x0.w * v0.w
            + x1.x * v1.x + x1.y * v1.y + x1.z * v1.z + x1.w * v1.w;

    // wave32 tree reduction
    #pragma unroll
    for (int off = 16; off > 0; off >>= 1)
        s += __shfl_xor(s, off, 32);

    if (lane == 0) ys[wave] = s;
}

// ---------------------------------------------------------------------------
extern "C" void kernel_launch(void* const* d_in, const int* in_sizes, int n_in,
                              void* d_out, int out_size, void* d_ws, size_t ws_size,
                              hipStream_t stream)
{
    const float* xss        = (const float*)d_in[0];
    const float* Wk         = (const float*)d_in[1];
    const float* Wc         = (const float*)d_in[2];
    const float* proj       = (const float*)d_in[3];
    const int*   seq_len    = (const int*)d_in[4];
    const int*   gather_idx = (const int*)d_in[5];
    float*       ys         = (float*)d_out;

    float* v_ws = (float*)d_ws;                // DIM_B * DIM_Q floats = 4 MB

    // Kernel 1: 4096 batches / 16 per block = 256 blocks
    ragged_pool_gemm_kernel<<<DIM_B / BPB, 256, 0, stream>>>(
        xss, Wk, Wc, proj, seq_len, v_ws);

    // Kernel 2: one wave (32 lanes) per valid token; 8 waves per block
    const int total = in_sizes[5];             // number of valid tokens
    const int blocks = (total + 7) / 8;
    token_dot_kernel<<<blocks, 256, 0, stream>>>(
        xss, v_ws, gather_idx, ys, total);
}